// GNN_76897094467963
// MI455X (gfx1250) — compile-verified
//
#include <hip/hip_runtime.h>
#include <hip/hip_bf16.h>
#include <math.h>

#define NNODES 100000
#define NEDGES 1600000
#define CDIM   64
#define NEG_SLOPE 0.2f

typedef __attribute__((ext_vector_type(2))) float v2f;
typedef __attribute__((ext_vector_type(8))) float v8f;

// ---------------------------------------------------------------------------
// init: agg[N*64]=0, m[N]=-inf, denom[N]=0
// ---------------------------------------------------------------------------
__global__ void init_bufs(float* __restrict__ agg, float* __restrict__ mmax,
                          float* __restrict__ denom, int n) {
    int i = blockIdx.x * blockDim.x + threadIdx.x;
    if (i < n * CDIM) agg[i] = 0.0f;
    if (i < n) {
        mmax[i]  = __int_as_float(0xFF800000); // -inf
        denom[i] = 0.0f;
    }
}

// ---------------------------------------------------------------------------
// wd[i] = sum_j W_dst[i,j] * att_dst[j]   (64 threads)
// ---------------------------------------------------------------------------
__global__ void gemv_wd(const float* __restrict__ W, const float* __restrict__ att,
                        float* __restrict__ wd) {
    int i = threadIdx.x;
    if (i >= CDIM) return;
    float s = 0.0f;
    #pragma unroll
    for (int j = 0; j < CDIM; ++j) s += W[i * CDIM + j] * att[j];
    wd[i] = s;
}

// ---------------------------------------------------------------------------
// WMMA GEMM: C[N,64] = A[N,64] @ W[64,64] (+bias)  — full fp32 via
// V_WMMA_F32_16X16X4_F32.  4 waves per block, each wave owns a 16-row tile
// and all four 16-col tiles; K=64 in 16 steps of K=4.
// ---------------------------------------------------------------------------
__global__ __launch_bounds__(128) void gemm64_wmma(
    const float* __restrict__ A, const float* __restrict__ W,
    const float* __restrict__ bias, float* __restrict__ C,
    int nrows, int has_bias)
{
    __shared__ float Ws[CDIM * CDIM];
    for (int i = threadIdx.x; i < CDIM * CDIM; i += blockDim.x) Ws[i] = W[i];
    __syncthreads();

    const int wave = threadIdx.x >> 5;
    const int lane = threadIdx.x & 31;
    const int tile = blockIdx.x * 4 + wave;           // 16-row tile
    if (tile * 16 >= nrows) return;                   // wave-uniform exit

    const int row0 = tile * 16;
    const int m    = lane & 15;
    const int hi   = lane >> 4;                       // half-wave select
    const float* __restrict__ Arow = A + (size_t)(row0 + m) * CDIM;

    v8f acc0 = {}, acc1 = {}, acc2 = {}, acc3 = {};

    #pragma unroll
    for (int k = 0; k < CDIM; k += 4) {
        // A 16x4 fp32 fragment: VGPR0 = K(k+2*hi), VGPR1 = K(k+2*hi+1)
        v2f a;
        a.x = Arow[k + 2 * hi];
        a.y = Arow[k + 2 * hi + 1];
        // B 4x16 fp32 fragments per column tile (same K striping as A)
        v2f b0, b1, b2, b3;
        const int kr0 = (k + 2 * hi) * CDIM;
        const int kr1 = (k + 2 * hi + 1) * CDIM;
        b0.x = Ws[kr0 + m +  0]; b0.y = Ws[kr1 + m +  0];
        b1.x = Ws[kr0 + m + 16]; b1.y = Ws[kr1 + m + 16];
        b2.x = Ws[kr0 + m + 32]; b2.y = Ws[kr1 + m + 32];
        b3.x = Ws[kr0 + m + 48]; b3.y = Ws[kr1 + m + 48];
        acc0 = __builtin_amdgcn_wmma_f32_16x16x4_f32(false, a, false, b0, (short)0, acc0, false, false);
        acc1 = __builtin_amdgcn_wmma_f32_16x16x4_f32(false, a, false, b1, (short)0, acc1, false, false);
        acc2 = __builtin_amdgcn_wmma_f32_16x16x4_f32(false, a, false, b2, (short)0, acc2, false, false);
        acc3 = __builtin_amdgcn_wmma_f32_16x16x4_f32(false, a, false, b3, (short)0, acc3, false, false);
    }

    // C/D layout: lane n = (lane&15), VGPR j holds row m = j + 8*hi
    v8f accs[4] = { acc0, acc1, acc2, acc3 };
    #pragma unroll
    for (int t = 0; t < 4; ++t) {
        const int n  = t * 16 + m;
        const float bv = has_bias ? bias[n] : 0.0f;
        #pragma unroll
        for (int j = 0; j < 8; ++j) {
            const int mm = j + 8 * hi;
            C[(size_t)(row0 + mm) * CDIM + n] = accs[t][j] + bv;
        }
    }
}

// ---------------------------------------------------------------------------
// per-node attention logits: a_s[i] = <xs[i,:], att_src>, a_d[i] = <xin[i,:], wd>
// ---------------------------------------------------------------------------
__global__ void node_dots(const float* __restrict__ xs, const float* __restrict__ xin,
                          const float* __restrict__ att_src, const float* __restrict__ wd,
                          float* __restrict__ a_s, float* __restrict__ a_d, int n) {
    int i = blockIdx.x * blockDim.x + threadIdx.x;
    if (i >= n) return;
    const float4* __restrict__ xsr = (const float4*)(xs  + (size_t)i * CDIM);
    const float4* __restrict__ xir = (const float4*)(xin + (size_t)i * CDIM);
    const float4* __restrict__ av  = (const float4*)att_src;
    const float4* __restrict__ wv  = (const float4*)wd;
    float s0 = 0.0f, s1 = 0.0f;
    #pragma unroll
    for (int j = 0; j < CDIM / 4; ++j) {
        float4 v = xsr[j], w = av[j];
        s0 += v.x * w.x + v.y * w.y + v.z * w.z + v.w * w.w;
        float4 u = xir[j], q = wv[j];
        s1 += u.x * q.x + u.y * q.y + u.z * q.z + u.w * q.w;
    }
    a_s[i] = s0;
    a_d[i] = s1;
}

// ---------------------------------------------------------------------------
// edge pass 1: e = leaky_relu(a_s[src]+a_d[dst]); segment-max into m[dst]
// ---------------------------------------------------------------------------
__device__ __forceinline__ void atomicMaxF(float* addr, float val) {
    if (val >= 0.0f) atomicMax((int*)addr, __float_as_int(val));
    else             atomicMin((unsigned int*)addr, __float_as_uint(val));
}

__global__ void edge_pass1(const long long* __restrict__ src, const long long* __restrict__ dst,
                           const float* __restrict__ a_s, const float* __restrict__ a_d,
                           float* __restrict__ eval, float* __restrict__ mmax, int nE) {
    int e = blockIdx.x * blockDim.x + threadIdx.x;
    if (e >= nE) return;
    int s = (int)src[e], d = (int)dst[e];
    float v = a_s[s] + a_d[d];
    v = v > 0.0f ? v : NEG_SLOPE * v;
    eval[e] = v;
    atomicMaxF(mmax + d, v);
}

// ---------------------------------------------------------------------------
// edge pass 2: ex = exp(e - m[dst]); denom[dst] += ex  (ex stored in place)
// ---------------------------------------------------------------------------
__global__ void edge_pass2(const long long* __restrict__ dst,
                           float* __restrict__ eval, const float* __restrict__ mmax,
                           float* __restrict__ denom, int nE) {
    int e = blockIdx.x * blockDim.x + threadIdx.x;
    if (e >= nE) return;
    int d = (int)dst[e];
    float ex = expf(eval[e] - mmax[d]);
    eval[e] = ex;
    atomicAdd(denom + d, ex);
}

// ---------------------------------------------------------------------------
// edge pass 3: agg[dst,:] += (ex/denom[dst]) * xs[src,:]
// 16 threads per edge, float4 gather, scalar f32 atomic adds.
// ---------------------------------------------------------------------------
__global__ void edge_scatter(const long long* __restrict__ src, const long long* __restrict__ dst,
                             const float* __restrict__ eval, const float* __restrict__ denom,
                             const float* __restrict__ xs, float* __restrict__ agg, int nE) {
    int t = blockIdx.x * blockDim.x + threadIdx.x;
    int e = t >> 4;
    if (e >= nE) return;
    int c = (t & 15) * 4;
    int s = (int)src[e], d = (int)dst[e];
    float alpha = eval[e] / denom[d];
    float4 v = *(const float4*)(xs + (size_t)s * CDIM + c);
    float* out = agg + (size_t)d * CDIM + c;
    atomicAdd(out + 0, alpha * v.x);
    atomicAdd(out + 1, alpha * v.y);
    atomicAdd(out + 2, alpha * v.z);
    atomicAdd(out + 3, alpha * v.w);
}

// ---------------------------------------------------------------------------
// layer-1 finalize: h = agg + lin + b_conv
// ---------------------------------------------------------------------------
__global__ void finalize_add(const float* __restrict__ agg, const float* __restrict__ lin,
                             const float* __restrict__ bconv, float* __restrict__ h, int total) {
    int i = blockIdx.x * blockDim.x + threadIdx.x;
    if (i >= total) return;
    h[i] = agg[i] + lin[i] + bconv[i & (CDIM - 1)];
}

// ---------------------------------------------------------------------------
// layer-2 finalize fused with row L2-normalize. One wave32 per node.
// ---------------------------------------------------------------------------
__global__ __launch_bounds__(32) void finalize_norm(const float* __restrict__ agg,
                                                    const float* __restrict__ lin,
                                                    const float* __restrict__ bconv,
                                                    float* __restrict__ out, int n) {
    int node = blockIdx.x;
    if (node >= n) return;
    int lane = threadIdx.x;
    size_t base = (size_t)node * CDIM;
    float v0 = agg[base + lane]      + lin[base + lane]      + bconv[lane];
    float v1 = agg[base + lane + 32] + lin[base + lane + 32] + bconv[lane + 32];
    float ss = v0 * v0 + v1 * v1;
    #pragma unroll
    for (int off = 16; off > 0; off >>= 1) ss += __shfl_xor(ss, off, 32);
    float norm = fmaxf(sqrtf(ss), 1e-12f);
    out[base + lane]      = v0 / norm;
    out[base + lane + 32] = v1 / norm;
}

// ---------------------------------------------------------------------------
// host: one GAT layer
// ---------------------------------------------------------------------------
static void run_gat_layer(const float* xin, const long long* src, const long long* dst,
                          const float* W_src, const float* W_dst,
                          const float* att_src, const float* att_dst,
                          const float* b_conv, const float* W_lin, const float* b_lin,
                          float* xs, float* lin, float* agg,
                          float* a_s, float* a_d, float* mmax, float* denom,
                          float* eval, float* wd,
                          float* out, int is_final, hipStream_t stream) {
    const int NC = NNODES * CDIM;
    init_bufs<<<(NC + 255) / 256, 256, 0, stream>>>(agg, mmax, denom, NNODES);
    gemv_wd<<<1, 64, 0, stream>>>(W_dst, att_dst, wd);
    gemm64_wmma<<<(NNODES / 16 + 3) / 4, 128, 0, stream>>>(xin, W_src, nullptr, xs, NNODES, 0);
    gemm64_wmma<<<(NNODES / 16 + 3) / 4, 128, 0, stream>>>(xin, W_lin, b_lin, lin, NNODES, 1);
    node_dots<<<(NNODES + 255) / 256, 256, 0, stream>>>(xs, xin, att_src, wd, a_s, a_d, NNODES);
    edge_pass1<<<(NEDGES + 255) / 256, 256, 0, stream>>>(src, dst, a_s, a_d, eval, mmax, NEDGES);
    edge_pass2<<<(NEDGES + 255) / 256, 256, 0, stream>>>(dst, eval, mmax, denom, NEDGES);
    edge_scatter<<<(NEDGES * 16 + 255) / 256, 256, 0, stream>>>(src, dst, eval, denom, xs, agg, NEDGES);
    if (is_final) {
        finalize_norm<<<NNODES, 32, 0, stream>>>(agg, lin, b_conv, out, NNODES);
    } else {
        finalize_add<<<(NC + 255) / 256, 256, 0, stream>>>(agg, lin, b_conv, out, NC);
    }
}

extern "C" void kernel_launch(void* const* d_in, const int* in_sizes, int n_in,
                              void* d_out, int out_size, void* d_ws, size_t ws_size,
                              hipStream_t stream) {
    const float*     x        = (const float*)d_in[0];
    const long long* eidx     = (const long long*)d_in[1];
    const float*     W_src1   = (const float*)d_in[2];
    const float*     W_dst1   = (const float*)d_in[3];
    const float*     att_src1 = (const float*)d_in[4];
    const float*     att_dst1 = (const float*)d_in[5];
    const float*     b_conv1  = (const float*)d_in[6];
    const float*     W_lin1   = (const float*)d_in[7];
    const float*     b_lin1   = (const float*)d_in[8];
    const float*     W_src2   = (const float*)d_in[9];
    const float*     W_dst2   = (const float*)d_in[10];
    const float*     att_src2 = (const float*)d_in[11];
    const float*     att_dst2 = (const float*)d_in[12];
    const float*     b_conv2  = (const float*)d_in[13];
    const float*     W_lin2   = (const float*)d_in[14];
    const float*     b_lin2   = (const float*)d_in[15];
    float*           out      = (float*)d_out;

    const long long* src = eidx;
    const long long* dst = eidx + NEDGES;

    // workspace carve-up (floats)
    const size_t NC = (size_t)NNODES * CDIM;
    float* w      = (float*)d_ws;
    float* xs     = w;              w += NC;
    float* lin    = w;              w += NC;
    float* h      = w;              w += NC;
    float* agg    = w;              w += NC;
    float* a_s    = w;              w += NNODES;
    float* a_d    = w;              w += NNODES;
    float* mmax   = w;              w += NNODES;
    float* denom  = w;              w += NNODES;
    float* eval   = w;              w += NEDGES;
    float* wd     = w;              w += CDIM;
    (void)ws_size; (void)n_in; (void)in_sizes; (void)out_size;

    // layer 1: h = GAT(x) + (x @ W_lin1 + b_lin1)
    run_gat_layer(x, src, dst, W_src1, W_dst1, att_src1, att_dst1, b_conv1,
                  W_lin1, b_lin1, xs, lin, agg, a_s, a_d, mmax, denom, eval, wd,
                  h, 0, stream);
    // layer 2 + normalize -> d_out
    run_gat_layer(h, src, dst, W_src2, W_dst2, att_src2, att_dst2, b_conv2,
                  W_lin2, b_lin2, xs, lin, agg, a_s, a_d, mmax, denom, eval, wd,
                  out, 1, stream);
}